// RowAttentionWithPairBias_42339787604439
// MI455X (gfx1250) — compile-verified
//
#include <hip/hip_runtime.h>
#include <math.h>

typedef float v2f __attribute__((ext_vector_type(2)));
typedef float v8f __attribute__((ext_vector_type(8)));

#define WMMA4(a, b, c) \
  __builtin_amdgcn_wmma_f32_16x16x4_f32(false, (a), false, (b), (short)0, (c), false, false)

// ---------------- constants ----------------
#define R     1024
#define CIN   256
#define NH    8
#define CH    32
#define CZ    128
#define HC    256          // NH*CH
#define QKVGW 1024         // 4*HC columns: [q | k | v | g]

// LDS bank-conflict swizzle for the score matrix (stride R floats)
#define SSW(row, col) ((col) ^ (((row) & 15) << 2))

__device__ __forceinline__ float wred_sum(float v) {
  v += __shfl_xor(v, 16, 32);
  v += __shfl_xor(v, 8, 32);
  v += __shfl_xor(v, 4, 32);
  v += __shfl_xor(v, 2, 32);
  v += __shfl_xor(v, 1, 32);
  return v;
}
__device__ __forceinline__ float g8_sum(float v) {
  v += __shfl_xor(v, 4, 32);
  v += __shfl_xor(v, 2, 32);
  v += __shfl_xor(v, 1, 32);
  return v;
}
__device__ __forceinline__ float g8_max(float v) {
  v = fmaxf(v, __shfl_xor(v, 4, 32));
  v = fmaxf(v, __shfl_xor(v, 2, 32));
  v = fmaxf(v, __shfl_xor(v, 1, 32));
  return v;
}

// ============ kernel 1: layernorm of m -> mln [1024,256] ============
__global__ void ln_m_kernel(const float* __restrict__ m,
                            const float* __restrict__ g,
                            const float* __restrict__ b,
                            float* __restrict__ mln) {
  int wave = threadIdx.x >> 5, lane = threadIdx.x & 31;
  int row = blockIdx.x * 8 + wave;
  const float* x = m + (size_t)row * CIN + lane * 8;
  float v[8];
  float4 p0 = *(const float4*)(x);
  float4 p1 = *(const float4*)(x + 4);
  v[0] = p0.x; v[1] = p0.y; v[2] = p0.z; v[3] = p0.w;
  v[4] = p1.x; v[5] = p1.y; v[6] = p1.z; v[7] = p1.w;
  float s = 0.f, sq = 0.f;
#pragma unroll
  for (int i = 0; i < 8; ++i) { s += v[i]; sq += v[i] * v[i]; }
  s = wred_sum(s); sq = wred_sum(sq);
  float mu = s * (1.f / CIN);
  float var = sq * (1.f / CIN) - mu * mu;
  float rs = rsqrtf(var + 1e-5f);
  float* o = mln + (size_t)row * CIN + lane * 8;
#pragma unroll
  for (int i = 0; i < 8; ++i) {
    int col = lane * 8 + i;
    o[i] = (v[i] - mu) * rs * g[col] + b[col];
  }
}

// ============ kernel 2: QKVG projection GEMM (1024x1024x256) =========
__global__ void proj_kernel(const float* __restrict__ mln,
                            const float* __restrict__ Wq,
                            const float* __restrict__ Wk,
                            const float* __restrict__ Wv,
                            const float* __restrict__ Wg,
                            const float* __restrict__ bg,
                            float* __restrict__ QKVG) {
  int lane = threadIdx.x;
  int tm = blockIdx.x, tn = blockIdx.y;
  int rm = tm * 16, cn = tn * 16;
  int mat = tn >> 4;                 // 0:q 1:k 2:v 3:g
  int ncol0 = (tn & 15) * 16;        // column base within selected matrix
  int lrow = lane & 15, hi = lane >> 4;
  const float* W = (mat == 0) ? Wq : (mat == 1) ? Wk : (mat == 2) ? Wv : Wg;

  v8f c = {};
  const float* ap = mln + (size_t)(rm + lrow) * CIN;
  for (int kb = 0; kb < CIN; kb += 4) {
    v2f a, bvec;
    int ka = kb + 2 * hi;
    a.x = ap[ka];
    a.y = ap[ka + 1];
    bvec.x = W[(size_t)(kb + hi) * CIN + ncol0 + lrow];
    bvec.y = W[(size_t)(kb + 2 + hi) * CIN + ncol0 + lrow];
    c = WMMA4(a, bvec, c);
  }
#pragma unroll
  for (int r = 0; r < 8; ++r) {
    int row = rm + r + 8 * hi;
    int col = cn + lrow;
    float val = c[r];
    if (mat == 0) val *= 0.17677669529663687f;      // 1/sqrt(CH)
    if (mat == 3) {
      val += bg[ncol0 + lrow];
      val = 1.0f / (1.0f + __expf(-val));           // sigmoid gate
    }
    QKVG[(size_t)row * QKVGW + col] = val;
  }
}

// ===== kernel 3: pair bias via WMMA with LN folded into the GEMM =====
// bias[h,i,j] = rs_ij * ( (z_ij . (g*Wz[:,h])) - mu_ij*128*S1n ) + S2[h]
// where S1[h]=sum_c g_c Wz[c,h], S2[h]=sum_c b_c Wz[c,h].
// One wave processes 16 tiles of 16 (i,j) pairs; K=128 -> 32 wmma/tile.
__global__ void bias_kernel(const float* __restrict__ z,
                            const float* __restrict__ g,
                            const float* __restrict__ b,
                            const float* __restrict__ Wz,
                            float* __restrict__ biasW) {
  int wave = threadIdx.x >> 5, lane = threadIdx.x & 31;
  int lrow = lane & 15, hi = lane >> 4;
  int n = lrow & 7;   // head column this lane owns (valid when lrow<8)

  // per-head LN fold constants (same for every pair)
  float S1 = 0.f, S2 = 0.f;
  for (int c = 0; c < CZ; ++c) {
    float wz = Wz[c * NH + n];
    S1 = fmaf(g[c], wz, S1);
    S2 = fmaf(b[c], wz, S2);
  }

  // B operand in registers: GW[k][col] = g[k]*Wz[k,col] (cols 8..15 are 0)
  float breg[64];
#pragma unroll
  for (int s = 0; s < 32; ++s) {
    int k0 = 4 * s + hi;
    int k1 = 4 * s + 2 + hi;
    breg[2 * s]     = (lrow < 8) ? g[k0] * Wz[k0 * NH + n] : 0.f;
    breg[2 * s + 1] = (lrow < 8) ? g[k1] * Wz[k1 * NH + n] : 0.f;
  }

  size_t tile0 = ((size_t)blockIdx.x * 8 + wave) * 16;
  for (int t = 0; t < 16; ++t) {
    size_t pairBase = (tile0 + t) * 16;          // 16 consecutive (i,j) pairs
    const float* zr = z + (pairBase + lrow) * CZ;
    __builtin_prefetch(zr + 16 * CZ, 0, 0);      // next tile's rows

    v8f c = {};
    float sum = 0.f, sq = 0.f;
#pragma unroll
    for (int s = 0; s < 32; ++s) {
      int k = 4 * s + 2 * hi;
      float2 av = *(const float2*)(zr + k);
      sum += av.x + av.y;
      sq = fmaf(av.x, av.x, sq);
      sq = fmaf(av.y, av.y, sq);
      v2f a, bb;
      a.x = av.x;          a.y = av.y;
      bb.x = breg[2 * s];  bb.y = breg[2 * s + 1];
      c = WMMA4(a, bb, c);
    }
    // lane l and l+16 together hold row (l&15): one butterfly completes sums
    sum += __shfl_xor(sum, 16, 32);
    sq  += __shfl_xor(sq, 16, 32);
    float mu = sum * (1.f / CZ);
    float var = sq * (1.f / CZ) - mu * mu;
    float rs = rsqrtf(var + 1e-5f);

#pragma unroll
    for (int r = 0; r < 8; ++r) {
      int srcLane = 8 * hi + r;                  // lane holding that row's stats
      float mu_r = __shfl(mu, srcLane, 32);
      float rs_r = __shfl(rs, srcLane, 32);
      float val = rs_r * (c[r] - mu_r * S1) + S2;
      if (lrow < 8) {                            // heads live in cols 0..7
        size_t pair = pairBase + 8 * hi + r;
        biasW[(size_t)lrow * ((size_t)R * R) + pair] = val;
      }
    }
  }
}

// ===== kernel 4: fused attention per (16-query tile, head) ===========
// block = 4 waves (128 threads). dynamic LDS = 64KB (16 x 1024 scores).
__global__ void attn_kernel(const float* __restrict__ QKVG,
                            const float* __restrict__ biasW,
                            float* __restrict__ OG) {
  extern __shared__ float smem[];     // [16][1024], XOR-swizzled columns
  int tid = threadIdx.x;
  int w = tid >> 5, lane = tid & 31;
  int qt = blockIdx.x, h = blockIdx.y;
  int hbase = h * CH;
  int lrow = lane & 15, hi = lane >> 4;

  // hoist q-tile A operand (16 queries x 32 dims) in registers
  float areg[16];
  const float* qrowp = QKVG + (size_t)(qt * 16 + lrow) * QKVGW + hbase;
#pragma unroll
  for (int s = 0; s < 8; ++s) {
    int ka = 4 * s + 2 * hi;
    areg[2 * s]     = qrowp[ka];
    areg[2 * s + 1] = qrowp[ka + 1];
  }

  // ---- phase 1: scores s = q @ k^T + bias into LDS ----
  for (int kt = w; kt < 64; kt += 4) {
    v8f c = {};
    const float* kp = QKVG + (size_t)(kt * 16 + lrow) * QKVGW + HC + hbase;
#pragma unroll
    for (int s = 0; s < 8; ++s) {
      v2f a, bvec;
      a.x = areg[2 * s];
      a.y = areg[2 * s + 1];
      int kd = 4 * s;
      bvec.x = kp[kd + hi];
      bvec.y = kp[kd + 2 + hi];
      c = WMMA4(a, bvec, c);
    }
#pragma unroll
    for (int r = 0; r < 8; ++r) {
      int row = r + 8 * hi;
      int col = kt * 16 + lrow;
      float v = c[r] +
          biasW[(size_t)h * ((size_t)R * R) + (size_t)(qt * 16 + row) * R + col];
      smem[row * R + SSW(row, col)] = v;
    }
  }
  __syncthreads();

  // ---- phase 2: softmax over each of 16 rows (8 threads / row) ----
  {
    int row = tid >> 3, sub = tid & 7;
    float* base = smem + row * R;
    float mx = -3.0e38f;
    for (int i = 0; i < 128; ++i) mx = fmaxf(mx, base[SSW(row, sub + 8 * i)]);
    mx = g8_max(mx);
    float sum = 0.f;
    for (int i = 0; i < 128; ++i) {
      int idx = SSW(row, sub + 8 * i);
      float e = __expf(base[idx] - mx);
      base[idx] = e;
      sum += e;
    }
    sum = g8_sum(sum);
    float inv = 1.0f / sum;
    for (int i = 0; i < 128; ++i) {
      int idx = SSW(row, sub + 8 * i);
      base[idx] *= inv;
    }
  }
  __syncthreads();

  // ---- phase 3: o = P @ V ; each wave owns a 256-wide K slice ----
  v8f acc[2];
  for (int nt = 0; nt < 2; ++nt) {
    v8f c = {};
    const float* vp = QKVG + 2 * HC + hbase + nt * 16 + lrow;
    for (int kk = 0; kk < 256; kk += 4) {
      int kb = w * 256 + kk;
      v2f a, bvec;
      int kaa = kb + 2 * hi;
      a.x = smem[lrow * R + SSW(lrow, kaa)];
      a.y = smem[lrow * R + SSW(lrow, kaa + 1)];
      bvec.x = vp[(size_t)(kb + hi) * QKVGW];
      bvec.y = vp[(size_t)(kb + 2 + hi) * QKVGW];
      c = WMMA4(a, bvec, c);
    }
    acc[nt] = c;
  }
  __syncthreads();   // done reading P; reuse LDS head for partials
  float* part = smem;  // [4 waves][16 rows][32 dims]
#pragma unroll
  for (int nt = 0; nt < 2; ++nt) {
#pragma unroll
    for (int r = 0; r < 8; ++r) {
      int row = r + 8 * hi;
      part[w * 512 + row * 32 + nt * 16 + lrow] = acc[nt][r];
    }
  }
  __syncthreads();

  // ---- phase 4: reduce wave partials, apply sigmoid gate, store ----
  for (int idx = tid; idx < 512; idx += 128) {
    int row = idx >> 5, d = idx & 31;
    float s = part[row * 32 + d] + part[512 + row * 32 + d] +
              part[1024 + row * 32 + d] + part[1536 + row * 32 + d];
    float gate = QKVG[(size_t)(qt * 16 + row) * QKVGW + 3 * HC + hbase + d];
    OG[(size_t)(qt * 16 + row) * HC + hbase + d] = s * gate;
  }
}

// ============ kernel 5: out = OG @ Wo + bo (1024x256x256) ============
__global__ void out_kernel(const float* __restrict__ OG,
                           const float* __restrict__ Wo,
                           const float* __restrict__ bo,
                           float* __restrict__ out) {
  int lane = threadIdx.x;
  int tm = blockIdx.x, tn = blockIdx.y;
  int rm = tm * 16, cn = tn * 16;
  int lrow = lane & 15, hi = lane >> 4;
  v8f c = {};
  const float* ap = OG + (size_t)(rm + lrow) * HC;
  for (int kb = 0; kb < HC; kb += 4) {
    v2f a, bvec;
    int ka = kb + 2 * hi;
    a.x = ap[ka];
    a.y = ap[ka + 1];
    bvec.x = Wo[(size_t)(kb + hi) * CIN + cn + lrow];
    bvec.y = Wo[(size_t)(kb + 2 + hi) * CIN + cn + lrow];
    c = WMMA4(a, bvec, c);
  }
#pragma unroll
  for (int r = 0; r < 8; ++r) {
    int row = rm + r + 8 * hi;
    int col = cn + lrow;
    out[(size_t)row * CIN + col] = c[r] + bo[col];
  }
}

// =====================================================================
extern "C" void kernel_launch(void* const* d_in, const int* in_sizes, int n_in,
                              void* d_out, int out_size, void* d_ws, size_t ws_size,
                              hipStream_t stream) {
  const float* m      = (const float*)d_in[0];
  const float* z      = (const float*)d_in[1];
  const float* ln_m_g = (const float*)d_in[2];
  const float* ln_m_b = (const float*)d_in[3];
  const float* ln_z_g = (const float*)d_in[4];
  const float* ln_z_b = (const float*)d_in[5];
  const float* Wz     = (const float*)d_in[6];
  const float* Wq     = (const float*)d_in[7];
  const float* Wk     = (const float*)d_in[8];
  const float* Wv     = (const float*)d_in[9];
  const float* Wg     = (const float*)d_in[10];
  const float* bg     = (const float*)d_in[11];
  const float* Wo     = (const float*)d_in[12];
  const float* bo     = (const float*)d_in[13];
  float* out = (float*)d_out;

  float* ws    = (float*)d_ws;
  float* mln   = ws;                                  // 1024*256
  float* QKVG  = mln + (size_t)R * CIN;               // 1024*1024
  float* biasW = QKVG + (size_t)R * QKVGW;            // 8*1024*1024
  float* OG    = biasW + (size_t)NH * R * R;          // 1024*256

  ln_m_kernel<<<dim3(R / 8), dim3(256), 0, stream>>>(m, ln_m_g, ln_m_b, mln);
  proj_kernel<<<dim3(64, 64), dim3(32), 0, stream>>>(mln, Wq, Wk, Wv, Wg, bg, QKVG);
  // 65536 M-tiles of 16 pairs; 16 tiles per wave; 8 waves per block
  bias_kernel<<<dim3(512), dim3(256), 0, stream>>>(z, ln_z_g, ln_z_b, Wz, biasW);
  attn_kernel<<<dim3(64, NH), dim3(128), 16 * R * sizeof(float), stream>>>(QKVG, biasW, OG);
  out_kernel<<<dim3(64, 16), dim3(32), 0, stream>>>(OG, Wo, bo, out);
}